// MultiScaleAttention_27908697489892
// MI455X (gfx1250) — compile-verified
//
#include <hip/hip_runtime.h>

// ---------------------------------------------------------------------------
// MultiScaleAttention for MI455X (gfx1250).
// fp32 WMMA (16x16x4) + Tensor Data Mover staging of weight tiles into LDS.
// attn tensor (356 MB, write-once) stored non-temporally to preserve L2.
// ---------------------------------------------------------------------------

typedef __attribute__((ext_vector_type(2))) float        v2f;
typedef __attribute__((ext_vector_type(8))) float        v8f;
typedef __attribute__((ext_vector_type(4))) unsigned int v4u;
typedef __attribute__((ext_vector_type(4))) int          v4i;
typedef __attribute__((ext_vector_type(8))) int          v8i;

#define WMMA_F32(a, b, c) \
  __builtin_amdgcn_wmma_f32_16x16x4_f32(false, (a), false, (b), (short)0, (c), false, false)

constexpr int B_  = 4;
constexpr int T_  = 512;
constexpr int QD_ = 256;
constexpr int H_  = 8;
constexpr int L_  = 4;
constexpr int DH_ = 32;
constexpr int S_  = 5440;   // 4096 + 1024 + 256 + 64

// LDS weight tile: 64 rows x 256 dwords, TDM-padded 2 dwords every 128 dwords
// -> row pitch 260 dwords (260 % 64 == 4 => conflict-free b64 fragment loads).
constexpr int WPITCH = 260;

// ---------------------------------------------------------------------------
// TDM: DMA one 64x256-float tile of W (row-major) into LDS offset 0 with
// hardware padding. Issued by one wave; completion via s_wait_tensorcnt.
// D# layout per CDNA5 ISA (groups 0/1; 2D tile).
// ---------------------------------------------------------------------------
__device__ __forceinline__ void tdm_load_w_tile(const float* wRow0)
{
  unsigned long long ga = (unsigned long long)(uintptr_t)wRow0;
  v4u g0;
  g0.x = 1u;                                             // count=1 (user descriptor)
  g0.y = 0u;                                             // lds_addr = 0 (dynamic LDS base)
  g0.z = (unsigned int)ga;                               // global_addr[31:0]
  g0.w = (unsigned int)((ga >> 32) & 0x01FFFFFFu) | (2u << 30);  // addr[56:32] | type=2

  v8i g1;
  g1[0] = (2 << 16) | (1 << 20) | (6 << 22) | (1 << 25); // data_size=4B, pad_en,
                                                         // interval=128dw, amount=2dw
  g1[1] = (256 << 16);                                   // tensor_dim0[15:0] = 256
  g1[2] = (64 << 16);                                    // tensor_dim1[15:0] = 64
  g1[3] = (256 << 16);                                   // tile_dim0 = 256
  g1[4] = 64;                                            // tile_dim1 = 64, tile_dim2 = 0
  g1[5] = 256;                                           // tensor_dim0_stride = 256
  g1[6] = 0;
  g1[7] = 0;

  v4i g2 = {0, 0, 0, 0};
  v4i g3 = {0, 0, 0, 0};
#if __clang_major__ >= 23
  v8i g4 = {0, 0, 0, 0, 0, 0, 0, 0};
  __builtin_amdgcn_tensor_load_to_lds(g0, g1, g2, g3, g4, 0);
#else
  __builtin_amdgcn_tensor_load_to_lds(g0, g1, g2, g3, 0);
#endif
}

// element k (0..255) of a TDM-padded LDS row
__device__ __forceinline__ int wpad(int k) { return k + ((k >> 7) << 1); }

// ---------------------------------------------------------------------------
// Projection GEMM: out[r, n] = in[r, :] . W[n, :] + bias[n]
//   MODE 0: q -> qs[b][h][t][d] (scaled);  MODE 1: k/v -> [b][h][s][d]
// Block tile M=64 (4 waves x 16), N=64, K=256. W tile staged via TDM.
// ---------------------------------------------------------------------------
template <int MODE, int RPB>
__global__ void __launch_bounds__(128)
proj_kernel(const float* __restrict__ in, const float* __restrict__ W,
            const float* __restrict__ bias, float* __restrict__ out, float scale)
{
  extern __shared__ float wt[];                       // [64][WPITCH]
  const int tid = threadIdx.x;
  const int n0  = blockIdx.y * 64;

  if (tid < 32) {                                     // wave 0 issues the DMA
    tdm_load_w_tile(W + (size_t)n0 * 256);
    __builtin_amdgcn_s_wait_tensorcnt(0);
  }
  __syncthreads();

  const int wave  = tid >> 5, lane = tid & 31;
  const int l16   = lane & 15;
  const int khalf = (lane >> 4) << 1;                 // 0 or 2
  const int m0    = blockIdx.x * 64 + wave * 16;
  const float* arow = in + (size_t)(m0 + l16) * 256 + khalf;

  v8f acc[4] = {};
  for (int k0 = 0; k0 < 256; k0 += 4) {
    v2f a = *(const v2f*)(arow + k0);                 // A[m][k0+khalf..+1]
    __builtin_prefetch(arow + k0 + 64, 0, 1);
    const int kadj = wpad(k0 + khalf);
#pragma unroll
    for (int nt = 0; nt < 4; ++nt) {
      // contiguous, 8B-aligned, conflict-free B fragment (one ds_load_b64)
      v2f b = *(const v2f*)(wt + (nt * 16 + l16) * WPITCH + kadj);
      acc[nt] = WMMA_F32(a, b, acc[nt]);
    }
  }

#pragma unroll
  for (int nt = 0; nt < 4; ++nt) {
    int   col = n0 + nt * 16 + l16;
    float bv  = bias[col];
    int   h   = col >> 5, d = col & 31;
#pragma unroll
    for (int r = 0; r < 8; ++r) {
      int gr = m0 + r + ((lane >> 4) << 3);           // C: M = r + 8*(lane/16)
      int bb = gr / RPB, ss = gr % RPB;               // compile-time divisor
      float val = acc[nt][r] + bv;
      size_t idx;
      if (MODE == 0) { val *= scale; idx = ((size_t)(bb * H_ + h) * T_ + ss) * DH_ + d; }
      else           {               idx = ((size_t)(bb * H_ + h) * S_ + ss) * DH_ + d; }
      out[idx] = val;
    }
  }
}

// ---------------------------------------------------------------------------
// Gate logits + softmax over L=4 (tiny). One wave per query row.
// ---------------------------------------------------------------------------
__global__ void __launch_bounds__(32)
gates_kernel(const float* __restrict__ q, const float* __restrict__ Wl,
             const float* __restrict__ bl, float* __restrict__ gates)
{
  __shared__ float lg[32];
  const int r = blockIdx.x, tid = threadIdx.x;
  const int bb = r / T_, tt = r % T_;
  const float* qr = q  + (size_t)r   * QD_;
  const float* wr = Wl + (size_t)tid * QD_;
  float acc = bl[tid];
  for (int kk = 0; kk < QD_; kk += 4)
    acc += qr[kk] * wr[kk] + qr[kk + 1] * wr[kk + 1] +
           qr[kk + 2] * wr[kk + 2] + qr[kk + 3] * wr[kk + 3];
  lg[tid] = acc;
  __syncthreads();
  if (tid < H_) {
    float m = lg[tid * 4];
#pragma unroll
    for (int l = 1; l < L_; ++l) m = fmaxf(m, lg[tid * 4 + l]);
    float e[L_], s = 0.f;
#pragma unroll
    for (int l = 0; l < L_; ++l) { e[l] = __expf(lg[tid * 4 + l] - m); s += e[l]; }
    float inv = 1.f / s;
#pragma unroll
    for (int l = 0; l < L_; ++l)
      gates[((size_t)(bb * H_ + tid) * T_ + tt) * L_ + l] = e[l] * inv;
  }
}

// ---------------------------------------------------------------------------
// Attention: one block (8 waves, 256 threads) per (b, h, 16-query tile).
// Score tile resident in LDS (pitch len+8). Softmax scans run conflict-free
// on 8 threads/row (first 128 threads); bulk passes use all 256 threads.
// attn written to HBM exactly once, non-temporally (preserve k/v in L2).
// ---------------------------------------------------------------------------
__global__ void __launch_bounds__(256)
attn_kernel(const float* __restrict__ qs, const float* __restrict__ ks,
            const float* __restrict__ vs, const float* __restrict__ gates,
            float* __restrict__ attn, float* __restrict__ hout)
{
  extern __shared__ float smem[];
  float* sc     = smem;                 // 16 * (4096+8)
  float* red    = sc + 16 * 4104;       // 128
  float* rowmax = red + 128;            // 16
  float* roww   = rowmax + 16;          // 16
  float* obuf   = roww + 16;            // 8 * 16 * 32

  const int LOFF[4] = {0, 4096, 5120, 5376};
  const int LLEN[4] = {4096, 1024, 256, 64};
  const int LLG[4]  = {12, 10, 8, 6};

  const int tid = threadIdx.x, wave = tid >> 5, lane = tid & 31;
  const int l16 = lane & 15, khalf = (lane >> 4) << 1;
  const int tq  = blockIdx.x & 31;
  const int bh  = blockIdx.x >> 5;      // b*H + h
  const int t0  = tq * 16;

  const size_t qbase = (size_t)bh * T_ + t0;

  // Preload 16x32 scaled-q tile as 8 WMMA A-fragments (VGPR-resident).
  const float* qrow = qs + (qbase + l16) * DH_ + khalf;
  v2f afrag[8];
#pragma unroll
  for (int kk = 0; kk < 8; ++kk) afrag[kk] = *(const v2f*)(qrow + kk * 4);

  const float* kbase    = ks + (size_t)bh * S_ * DH_;
  const float* vbase    = vs + (size_t)bh * S_ * DH_;
  float*       attnbase = attn + qbase * S_;

  const int rrow = tid >> 3, rsub = tid & 7;   // 8 threads per row (tid < 128)

  v8f oacc[2] = {};

#pragma unroll
  for (int lvl = 0; lvl < 4; ++lvl) {
    const int off = LOFF[lvl], len = LLEN[lvl], pitch = len + 8, lg2 = LLG[lvl];

    // ---- scores = q_tile @ k^T, key tiles strided over 8 waves ----
    for (int st = wave; st < (len >> 4); st += 8) {
      const int sb = off + st * 16;
      const float* krow = kbase + (size_t)(sb + l16) * DH_ + khalf;
      __builtin_prefetch(krow + 128 * DH_, 0, 1);
      v8f c = {};
#pragma unroll
      for (int kk = 0; kk < 8; ++kk) {
        v2f b = *(const v2f*)(krow + kk * 4);
        c = WMMA_F32(afrag[kk], b, c);
      }
#pragma unroll
      for (int r = 0; r < 8; ++r) {
        int m = r + ((lane >> 4) << 3);
        sc[m * pitch + st * 16 + l16] = c[r];
      }
    }
    __syncthreads();

    // ---- segmented softmax, gated (scans: first 128 threads, 8/row) ----
    if (tid < 128) {
      float mx = -3.0e38f;
      for (int j = rsub; j < len; j += 8) mx = fmaxf(mx, sc[rrow * pitch + j]);
      red[tid] = mx;
    }
    __syncthreads();
    if (tid < 128 && rsub == 0) {
      float m2 = red[tid];
#pragma unroll
      for (int i = 1; i < 8; ++i) m2 = fmaxf(m2, red[tid + i]);
      rowmax[rrow] = m2;
    }
    __syncthreads();
    if (tid < 128) {
      float rm = rowmax[rrow], sm = 0.f;
      for (int j = rsub; j < len; j += 8) {
        float e_ = __expf(sc[rrow * pitch + j] - rm);
        sc[rrow * pitch + j] = e_;
        sm += e_;
      }
      red[tid] = sm;
    }
    __syncthreads();
    if (tid < 128 && rsub == 0) {
      float s2 = 0.f;
#pragma unroll
      for (int i = 0; i < 8; ++i) s2 += red[tid + i];
      float gate = gates[(qbase + rrow) * L_ + lvl];
      roww[rrow] = gate / s2;
    }
    __syncthreads();

    // normalize in LDS + single coalesced NON-TEMPORAL global write of attn
    for (int e = tid; e < (len << 4); e += 256) {
      int r_ = e >> lg2, j = e & (len - 1);
      float a_ = sc[r_ * pitch + j] * roww[r_];
      sc[r_ * pitch + j] = a_;
      __builtin_nontemporal_store(a_, &attnbase[(size_t)r_ * S_ + off + j]);
    }
    __syncthreads();

    // ---- out += attn_level @ v_level (aligned b64 A-fragments from LDS) ----
    for (int kk = wave * 4; kk < len; kk += 32) {
      v2f a = *(const v2f*)(sc + l16 * pitch + kk + khalf);
      const float* vrow = vbase + (size_t)(off + kk + khalf) * DH_;
#pragma unroll
      for (int nt = 0; nt < 2; ++nt) {
        v2f b;
        b.x = vrow[nt * 16 + l16];
        b.y = vrow[DH_ + nt * 16 + l16];
        oacc[nt] = WMMA_F32(a, b, oacc[nt]);
      }
    }
    __syncthreads();   // sc reused by next level
  }

  // ---- cross-wave reduction of the 16x32 head output ----
#pragma unroll
  for (int nt = 0; nt < 2; ++nt)
#pragma unroll
    for (int r = 0; r < 8; ++r) {
      int m = r + ((lane >> 4) << 3), d = nt * 16 + l16;
      obuf[wave * 512 + m * 32 + d] = oacc[nt][r];
    }
  __syncthreads();
  for (int e = tid; e < 512; e += 256) {
    float s = 0.f;
#pragma unroll
    for (int w = 0; w < 8; ++w) s += obuf[w * 512 + e];
    int m = e >> 5, d = e & 31;
    hout[(qbase + m) * DH_ + d] = s;
  }
}

// ---------------------------------------------------------------------------
// Output projection: out[b,t,:] = concat_h(hout[b,h,t,:]) @ Wo^T + bo
// ---------------------------------------------------------------------------
__global__ void __launch_bounds__(128)
outproj_kernel(const float* __restrict__ hout, const float* __restrict__ Wo,
               const float* __restrict__ bo, float* __restrict__ out)
{
  extern __shared__ float wt[];
  const int tid = threadIdx.x;
  const int n0  = blockIdx.y * 64;

  if (tid < 32) {
    tdm_load_w_tile(Wo + (size_t)n0 * 256);
    __builtin_amdgcn_s_wait_tensorcnt(0);
  }
  __syncthreads();

  const int wave = tid >> 5, lane = tid & 31;
  const int l16 = lane & 15, khalf = (lane >> 4) << 1;
  const int m0 = blockIdx.x * 64 + wave * 16;
  const int gr = m0 + l16;
  const int bb = gr / T_, tt = gr % T_;
  const float* hbase = hout + ((size_t)bb * H_ * T_ + tt) * DH_;  // + h*T*DH + d

  v8f acc[4] = {};
  for (int k0 = 0; k0 < 256; k0 += 4) {
    int e0 = k0 + khalf;                                  // e = h*32 + d
    v2f a = *(const v2f*)(hbase + (size_t)(e0 >> 5) * T_ * DH_ + (e0 & 31));
    const int kadj = wpad(e0);
#pragma unroll
    for (int nt = 0; nt < 4; ++nt) {
      v2f b = *(const v2f*)(wt + (nt * 16 + l16) * WPITCH + kadj);
      acc[nt] = WMMA_F32(a, b, acc[nt]);
    }
  }

#pragma unroll
  for (int nt = 0; nt < 4; ++nt) {
    int   col = n0 + nt * 16 + l16;
    float bv  = bo[col];
#pragma unroll
    for (int r = 0; r < 8; ++r) {
      int row = m0 + r + ((lane >> 4) << 3);
      out[(size_t)row * QD_ + col] = acc[nt][r] + bv;
    }
  }
}

// ---------------------------------------------------------------------------
extern "C" void kernel_launch(void* const* d_in, const int* in_sizes, int n_in,
                              void* d_out, int out_size, void* d_ws, size_t ws_size,
                              hipStream_t stream)
{
  (void)in_sizes; (void)n_in; (void)out_size; (void)ws_size;

  const float* q  = (const float*)d_in[0];
  const float* k  = (const float*)d_in[1];
  const float* v  = (const float*)d_in[2];
  const float* Wq = (const float*)d_in[3];
  const float* bq = (const float*)d_in[4];
  const float* Wk = (const float*)d_in[5];
  const float* bk = (const float*)d_in[6];
  const float* Wv = (const float*)d_in[7];
  const float* bv = (const float*)d_in[8];
  const float* Wo = (const float*)d_in[9];
  const float* bo = (const float*)d_in[10];
  const float* Wl = (const float*)d_in[11];
  const float* bl = (const float*)d_in[12];
  // d_in[13] = k_spatial_shapes (static metadata, baked in as constants)

  float* ws    = (float*)d_ws;
  float* qs    = ws;                                    // B*H*T*DH
  float* ks    = qs    + (size_t)B_ * H_ * T_ * DH_;    // B*H*S*DH
  float* vs    = ks    + (size_t)B_ * H_ * S_ * DH_;    // B*H*S*DH
  float* gates = vs    + (size_t)B_ * H_ * S_ * DH_;    // B*H*T*L
  float* hout  = gates + (size_t)B_ * H_ * T_ * L_;     // B*H*T*DH

  float* outp = (float*)d_out;                          // [B,T,QD]
  float* attn = outp + (size_t)B_ * T_ * QD_;           // [B,H,T,S]

  const float  scale   = 0.17677669529663687f;          // DH^-0.5
  const size_t projLds = 64 * WPITCH * sizeof(float);   // 66,560 B
  const size_t attnLds = (16 * 4104 + 128 + 16 + 16 + 8 * 16 * 32) * sizeof(float);

  proj_kernel<0, T_><<<dim3(B_ * T_ / 64, 4), 128, projLds, stream>>>(q, Wq, bq, qs, scale);
  proj_kernel<1, S_><<<dim3(B_ * S_ / 64, 4), 128, projLds, stream>>>(k, Wk, bk, ks, 1.f);
  proj_kernel<1, S_><<<dim3(B_ * S_ / 64, 4), 128, projLds, stream>>>(v, Wv, bv, vs, 1.f);
  gates_kernel<<<dim3(B_ * T_), 32, 0, stream>>>(q, Wl, bl, gates);
  attn_kernel<<<dim3(B_ * H_ * (T_ / 16)), 256, attnLds, stream>>>(qs, ks, vs, gates, attn, hout);
  outproj_kernel<<<dim3(B_ * T_ / 64, 4), 128, projLds, stream>>>(hout, Wo, bo, outp);
}